// HeadwiseThreshold_37383395344632
// MI455X (gfx1250) — compile-verified
//
#include <hip/hip_runtime.h>
#include <hip/hip_bf16.h>

// Problem constants from the reference (fixed shapes).
#define B_ 32
#define H_ 12
#define N_ 784
#define M_ 784

typedef float v2f __attribute__((ext_vector_type(2)));
typedef float v8f __attribute__((ext_vector_type(8)));

// ---------------------------------------------------------------------------
// Kernel 1: per (b,h) block:
//   - load importance row + sim_threshold row into LDS
//   - cummax the threshold row (monotonic threshold along kept_num axis)
//   - rank[n] = #{j: imp[j] > imp[n]} + #{j<n: imp[j]==imp[n]}  (stable
//     descending double-argsort semantics)
//   - sorted_threshold[b,h,n] = cummax_thr[rank[n]]  -> written to out region 3
// ---------------------------------------------------------------------------
__global__ void __launch_bounds__(256)
rank_thresh_kernel(const float* __restrict__ imp,      // [B,H,N]
                   const float* __restrict__ thr_in,   // [H,N]
                   float* __restrict__ st_out)         // [B,H,N]
{
    __shared__ float s_imp[N_];
    __shared__ float s_thr[N_];

    const int bh = blockIdx.x;        // b*H + h
    const int h  = bh % H_;
    const int tid = threadIdx.x;

    const float* imp_row = imp + (size_t)bh * N_;
    const float* thr_row = thr_in + (size_t)h * N_;
    for (int i = tid; i < N_; i += 256) {
        s_imp[i] = imp_row[i];
        s_thr[i] = thr_row[i];
    }
    __syncthreads();

    // serial cummax (cheap: 784 iters, overlapped across 384 blocks)
    if (tid == 0) {
        float m = s_thr[0];
        for (int i = 1; i < N_; ++i) {
            float v = s_thr[i];
            m = (v > m) ? v : m;
            s_thr[i] = m;
        }
    }
    __syncthreads();

    for (int n = tid; n < N_; n += 256) {
        const float v = s_imp[n];
        int cnt = 0;
        for (int j = 0; j < N_; ++j) {
            const float u = s_imp[j];
            cnt += (int)(u > v) | ((int)(u == v) & (int)(j < n));
        }
        st_out[(size_t)bh * N_ + n] = s_thr[cnt];
    }
}

// ---------------------------------------------------------------------------
// Kernel 2 (heavy, streams 944 MB once): one block per (b,n).
// S[m] = sum_h w[h]*sim[b,h,n,m]; find max/argmax over m via WMMA f32 16x16x4:
//   A (16x4) rows = 16 consecutive m, K = h-chunk; B columns all = w[k].
// Three chained WMMAs cover h=0..11. C/D layout: lane (half,l) VGPR r holds
// S[m0 + 8*half + r] (replicated over l) -> register max/argmax.
// In-lane m is strictly increasing, so first-max semantics need only a strict
// '>' here; the (==, smaller-index) tie rule is applied at combine points.
// ---------------------------------------------------------------------------
__global__ void __launch_bounds__(128)
reuse_kernel(const float* __restrict__ sim,       // [B,H,N,M]
             const float* __restrict__ w,         // [H]
             const float* __restrict__ st,        // [B,H,N] (out region 3)
             float* __restrict__ out_score,       // [B,N]
             float* __restrict__ out_idx)         // [B,N] (index as float)
{
    const int n    = blockIdx.x;
    const int b    = blockIdx.y;
    const int tid  = threadIdx.x;
    const int wave = tid >> 5;
    const int lane = tid & 31;
    const int half = lane >> 4;

    const size_t NM = (size_t)N_ * M_;
    const float* rowBase = sim + ((size_t)b * H_) * NM + (size_t)n * M_;

    float bestV = -__builtin_inff();
    int   bestM = 0;

#if defined(__gfx1250__) && __has_builtin(__builtin_amdgcn_wmma_f32_16x16x4_f32)
    const int l = lane & 15;
    // B matrix: B[k, j] = w[k] for all j. VGPR0 lanes0-15 K=0 / lanes16-31 K=2,
    // VGPR1 lanes0-15 K=1 / lanes16-31 K=3 (per k-chunk base 4c).
    v2f b0, b1, b2;
    b0.x = w[0 + 2 * half]; b0.y = w[1 + 2 * half];
    b1.x = w[4 + 2 * half]; b1.y = w[5 + 2 * half];
    b2.x = w[8 + 2 * half]; b2.y = w[9 + 2 * half];

    // Per-chunk lane base pointers: h = 4c + 2*half (+1 via offset NM),
    // m = 16*t + l. Keeps every load's byte offset within the 24-bit imm.
    const float* p0 = rowBase + (size_t)(2 * half) * NM + l;          // h 0/2
    const float* p1 = p0 + 4 * NM;                                    // h 4/6
    const float* p2 = p0 + 8 * NM;                                    // h 8/10

    const int TILES = M_ / 16;   // 49, exact (no tail)
    for (int t = wave; t < TILES; t += 4) {
        const int moff = t * 16;
        v2f a0, a1, a2;
        a0.x = __builtin_nontemporal_load(p0 + moff);
        a0.y = __builtin_nontemporal_load(p0 + moff + NM);
        a1.x = __builtin_nontemporal_load(p1 + moff);
        a1.y = __builtin_nontemporal_load(p1 + moff + NM);
        a2.x = __builtin_nontemporal_load(p2 + moff);
        a2.y = __builtin_nontemporal_load(p2 + moff + NM);

        v8f acc = {0.f, 0.f, 0.f, 0.f, 0.f, 0.f, 0.f, 0.f};
        acc = __builtin_amdgcn_wmma_f32_16x16x4_f32(false, a0, false, b0,
                                                    (short)0, acc, false, false);
        acc = __builtin_amdgcn_wmma_f32_16x16x4_f32(false, a1, false, b1,
                                                    (short)0, acc, false, false);
        acc = __builtin_amdgcn_wmma_f32_16x16x4_f32(false, a2, false, b2,
                                                    (short)0, acc, false, false);

        const int mBase = moff + 8 * half;
#pragma unroll
        for (int r = 0; r < 8; ++r) {
            const float v   = acc[r];
            const bool take = v > bestV;                 // branchless selects
            bestM = take ? (mBase + r) : bestM;
            bestV = take ? v : bestV;
        }
    }
#else
    // Scalar fallback: coalesced strided m, 12 FMAs per element.
    float wr[H_];
#pragma unroll
    for (int h = 0; h < H_; ++h) wr[h] = w[h];
    for (int m = tid; m < M_; m += 128) {
        float s = 0.f;
#pragma unroll
        for (int h = 0; h < H_; ++h)
            s = fmaf(__builtin_nontemporal_load(rowBase + (size_t)h * NM + m),
                     wr[h], s);
        const bool take = s > bestV;
        bestM = take ? m : bestM;
        bestV = take ? s : bestV;
    }
#endif

    // wave32 butterfly reduction, branchless, first-max-index tie rule
    for (int off = 16; off >= 1; off >>= 1) {
        const float ov  = __shfl_xor(bestV, off, 32);
        const int   oi  = __shfl_xor(bestM, off, 32);
        const bool take = (ov > bestV) | ((ov == bestV) & (oi < bestM));
        bestV = take ? ov : bestV;
        bestM = take ? oi : bestM;
    }

    __shared__ float sV[4];
    __shared__ int   sI[4];
    if (lane == 0) { sV[wave] = bestV; sI[wave] = bestM; }
    __syncthreads();

    if (tid == 0) {
        float v  = sV[0];
        int   mi = sI[0];
#pragma unroll
        for (int i = 1; i < 4; ++i) {
            const float ov  = sV[i];
            const int   oi  = sI[i];
            const bool take = (ov > v) | ((ov == v) & (oi < mi));
            v  = take ? ov : v;
            mi = take ? oi : mi;
        }

        // t[b,n] = sum_h w[h] * sorted_threshold[b,h,n]
        float tval = 0.f;
        const float* stp = st + ((size_t)b * H_) * N_ + n;
#pragma unroll
        for (int h = 0; h < H_; ++h) tval = fmaf(w[h], stp[(size_t)h * N_], tval);

        const size_t o = (size_t)b * N_ + n;
        out_score[o] = 10.0f * tanhf(v - tval);
        out_idx[o]   = (float)mi;
    }
}

// ---------------------------------------------------------------------------
// Launch: inputs are (importance, similarity, compressed_map[unused],
// sim_threshold, linear_w). Outputs flat-concatenated:
//   [0, B*N)            most_similar_score  [B,N,1]
//   [B*N, 2*B*N)        most_similar_idx    [B,N]   (value-cast to float)
//   [2*B*N, ...)        sorted_threshold    [B,H,N,1]
// ---------------------------------------------------------------------------
extern "C" void kernel_launch(void* const* d_in, const int* in_sizes, int n_in,
                              void* d_out, int out_size, void* d_ws, size_t ws_size,
                              hipStream_t stream) {
    (void)in_sizes; (void)n_in; (void)out_size; (void)d_ws; (void)ws_size;

    const float* imp    = (const float*)d_in[0];
    const float* sim    = (const float*)d_in[1];
    const float* thr_in = (const float*)d_in[3];
    const float* w      = (const float*)d_in[4];

    float* out       = (float*)d_out;
    float* out_score = out;                                 // [B,N]
    float* out_idx   = out + (size_t)B_ * N_;               // [B,N]
    float* out_st    = out + (size_t)2 * B_ * N_;           // [B,H,N]

    rank_thresh_kernel<<<B_ * H_, 256, 0, stream>>>(imp, thr_in, out_st);

    dim3 grid(N_, B_);
    reuse_kernel<<<grid, 128, 0, stream>>>(sim, w, out_st, out_score, out_idx);
}